// VariationalGraphAE_16853451670121
// MI455X (gfx1250) — compile-verified
//
#include <hip/hip_runtime.h>
#include <hip/hip_bf16.h>

typedef __attribute__((ext_vector_type(2))) float v2f;
typedef __attribute__((ext_vector_type(8))) float v8f;

// ---------------------------------------------------------------- utilities

__global__ void zero_f_kernel(float* __restrict__ p, long long n) {
    long long i = (long long)blockIdx.x * blockDim.x + threadIdx.x;
    if (i < n) p[i] = 0.0f;
}

// Per-edge degree-2 closed B-spline basis + tap indices + in-degree count.
__global__ void basis_deg_kernel(const float* __restrict__ pseudo,
                                 const int* __restrict__ ei,   // [2,E]: src row 0, dst row 1
                                 float* __restrict__ basis,    // [E,3]
                                 int* __restrict__ widx,       // [E,3]
                                 float* __restrict__ deg,      // [N]
                                 int E) {
    int e = blockIdx.x * blockDim.x + threadIdx.x;
    if (e >= E) return;
    float v  = pseudo[e] * 3.0f;
    float fl = floorf(v);
    float f  = v - fl;
    basis[e * 3 + 0] = 0.5f * f * f - f + 0.5f;
    basis[e * 3 + 1] = -f * f + f + 0.5f;
    basis[e * 3 + 2] = 0.5f * f * f;
    int b0 = (int)fl % 3;
    if (b0 < 0) b0 += 3;
    widx[e * 3 + 0] = b0;
    widx[e * 3 + 1] = (b0 + 1) % 3;
    widx[e * 3 + 2] = (b0 + 2) % 3;
    atomicAdd(&deg[ei[E + e]], 1.0f);
}

__global__ void deg_inv_kernel(float* __restrict__ deg, int n) {
    int i = blockIdx.x * blockDim.x + threadIdx.x;
    if (i < n) deg[i] = 1.0f / fmaxf(deg[i], 1.0f);
}

// Repack spline weight [K=3][cin][cout] -> [cin][3*cout], wp[i][k*cout+o] = w[k][i][o]
__global__ void repack_w_kernel(const float* __restrict__ w, float* __restrict__ wp,
                                int cin, int cout) {
    int idx = blockIdx.x * blockDim.x + threadIdx.x;
    int total = 3 * cin * cout;
    if (idx >= total) return;
    int k   = idx / (cin * cout);
    int rem = idx - k * cin * cout;
    int i   = rem / cout;
    int o   = rem - i * cout;
    wp[(size_t)i * (3 * cout) + k * cout + o] = w[idx];
}

// ---------------------------------------------------------------- WMMA GEMM
// Y[nrows, cout] = X[nrows, cin] * W[cin, cout]  (+ bias, optional ReLU)
// cin multiple of 4 (64/128), cout multiple of 16. Block = 128 threads = 4 waves,
// each wave computes one 16x16 tile; block covers 64 rows x 16 cols.
// Fragment layouts per CDNA5 ISA §7.12.2:
//   A 16x4 f32 : lanes 0-15 row M=lane, VGPR{0,1}=K{0,1}; lanes 16-31 -> K{2,3}
//   B 4x16 f32 : lane%16 = column N; lane-half splits K the same way
//   C/D 16x16  : VGPR r -> row r (lanes 0-15) / row r+8 (lanes 16-31), N = lane%16
#define GEMM_ROWS 64
#define GEMM_THREADS 128

__global__ __launch_bounds__(GEMM_THREADS)
void gemm_wmma_f32_kernel(const float* __restrict__ X, const float* __restrict__ W,
                          const float* __restrict__ bias, float* __restrict__ Y,
                          int nrows, int cin, int cout, int dorelu) {
    __shared__ float sA[GEMM_ROWS * 128];   // up to cin=128
    __shared__ float sB[128 * 16];

    const int tid  = threadIdx.x;
    const int lane = tid & 31;
    const int wv   = tid >> 5;        // wave 0..3
    const int half = lane >> 4;       // 0 | 1 (K-half select)
    const int m    = lane & 15;       // row (A) / col (B,C)

    const int row0 = blockIdx.x * GEMM_ROWS;
    const int col0 = blockIdx.y * 16;

    // stage A tile [GEMM_ROWS x cin] (coalesced; zero-pad OOB rows)
    for (int idx = tid; idx < GEMM_ROWS * cin; idx += GEMM_THREADS) {
        int r = idx / cin, c = idx - r * cin;
        int gr = row0 + r;
        sA[idx] = (gr < nrows) ? X[(size_t)gr * cin + c] : 0.0f;
    }
    // stage B tile [cin x 16]
    for (int idx = tid; idx < cin * 16; idx += GEMM_THREADS) {
        int k = idx >> 4, n = idx & 15;
        sB[idx] = W[(size_t)k * cout + col0 + n];
    }
    __syncthreads();

    v8f acc = {};
    const float* aRow = &sA[(wv * 16 + m) * cin];
    for (int k0 = 0; k0 < cin; k0 += 4) {
        int kb = k0 + half * 2;
        v2f a, b;
        a.x = aRow[kb];
        a.y = aRow[kb + 1];
        b.x = sB[kb * 16 + m];
        b.y = sB[(kb + 1) * 16 + m];
        acc = __builtin_amdgcn_wmma_f32_16x16x4_f32(
            /*neg_a=*/false, a, /*neg_b=*/false, b,
            /*c_mod=*/(short)0, acc, /*reuse_a=*/false, /*reuse_b=*/false);
    }

    float bv = bias ? bias[col0 + m] : 0.0f;
#pragma unroll
    for (int r = 0; r < 8; ++r) {
        int gr = row0 + wv * 16 + (half ? (r + 8) : r);
        if (gr < nrows) {
            float v = acc[r] + bv;
            if (dorelu) v = fmaxf(v, 0.0f);
            Y[(size_t)gr * cout + col0 + m] = v;
        }
    }
}

// ---------------------------------------------------------------- edge scatter
// out[dst, c] += sum_s basis[e,s] * y[src, widx[e,s]*cout + c]
__global__ void scatter_edges_kernel(const float* __restrict__ y,
                                     const float* __restrict__ basis,
                                     const int* __restrict__ widx,
                                     const int* __restrict__ ei,
                                     float* __restrict__ out,
                                     int E, int cout, long long total) {
    long long idx = (long long)blockIdx.x * blockDim.x + threadIdx.x;
    if (idx >= total) return;
    int e = (int)(idx / cout);
    int c = (int)(idx - (long long)e * cout);
    int s = ei[e];
    int d = ei[E + e];
    const float* yrow = y + (size_t)s * (3 * cout);
    float acc = 0.0f;
#pragma unroll
    for (int t = 0; t < 3; ++t) {
        float b = basis[e * 3 + t];
        int w   = widx[e * 3 + t];
        acc += b * yrow[w * cout + c];
    }
    atomicAdd(&out[(size_t)d * cout + c], acc);
}

__global__ void norm_deg_kernel(float* __restrict__ out, const float* __restrict__ deginv,
                                int cout, long long total) {
    long long idx = (long long)blockIdx.x * blockDim.x + threadIdx.x;
    if (idx >= total) return;
    out[idx] *= deginv[idx / cout];
}

// z = mu + eps * exp(0.5 * logvar)
__global__ void reparam_kernel(const float* __restrict__ mu, const float* __restrict__ logvar,
                               const float* __restrict__ eps, float* __restrict__ z,
                               long long n) {
    long long i = (long long)blockIdx.x * blockDim.x + threadIdx.x;
    if (i < n) z[i] = mu[i] + eps[i] * __expf(0.5f * logvar[i]);
}

// ---------------------------------------------------------------- launch

extern "C" void kernel_launch(void* const* d_in, const int* in_sizes, int n_in,
                              void* d_out, int out_size, void* d_ws, size_t ws_size,
                              hipStream_t stream) {
    const float* x      = (const float*)d_in[0];
    const float* pseudo = (const float*)d_in[1];
    const float* w1     = (const float*)d_in[2];
    const float* w2     = (const float*)d_in[3];
    const float* mu_w   = (const float*)d_in[4];
    const float* lv_w   = (const float*)d_in[5];
    const float* d1_w   = (const float*)d_in[6];   // [64,128] already [in,out]
    const float* d1_b   = (const float*)d_in[7];
    const float* d2_w   = (const float*)d_in[8];   // [128,64]
    const float* d2_b   = (const float*)d_in[9];
    const float* eps    = (const float*)d_in[10];
    const int*   ei     = (const int*)d_in[11];    // [2,E]

    const int IN_CH = 64, HID = 128, LAT = 64;
    const int N = in_sizes[0] / IN_CH;
    const int E = in_sizes[1];

    // workspace layout (floats)
    float* wsf   = (float*)d_ws;
    float* deg   = wsf;                               // N (then holds deg^-1)
    float* basis = deg + N;                           // 3E
    int*   widx  = (int*)(basis + (size_t)3 * E);     // 3E
    float* ybuf  = (float*)(widx + (size_t)3 * E);    // N*384 (max 3*HID)
    float* h1    = ybuf + (size_t)N * 384;            // N*128
    float* h2    = h1 + (size_t)N * 128;              // N*64
    float* wp    = h2 + (size_t)N * 64;               // 3*128*64 = 24576
    float* z     = h1;                                // reuse (h1 dead after sconv2)
    float* hdec  = ybuf;                              // reuse (ybuf dead after last sconv)

    float* rec    = (float*)d_out;                    // [N,64]
    float* mu     = rec + (size_t)N * IN_CH;          // [N,64]
    float* logvar = mu + (size_t)N * LAT;             // [N,64]

    auto blocks = [](long long n) { return dim3((unsigned)((n + 255) / 256)); };

    // degree + basis
    zero_f_kernel<<<blocks(N), 256, 0, stream>>>(deg, N);
    basis_deg_kernel<<<blocks(E), 256, 0, stream>>>(pseudo, ei, basis, widx, deg, E);
    deg_inv_kernel<<<blocks(N), 256, 0, stream>>>(deg, N);

    // one spline conv: hout[N,cout] = norm * scatter(basis * (hin @ w)[src, tap])
    auto sconv = [&](const float* hin, const float* w, int cin, int cout, float* hout) {
        long long t = 3LL * cin * cout;
        repack_w_kernel<<<blocks(t), 256, 0, stream>>>(w, wp, cin, cout);
        dim3 g((N + GEMM_ROWS - 1) / GEMM_ROWS, (3 * cout) / 16);
        gemm_wmma_f32_kernel<<<g, GEMM_THREADS, 0, stream>>>(hin, wp, nullptr, ybuf,
                                                             N, cin, 3 * cout, 0);
        long long nc = (long long)N * cout;
        zero_f_kernel<<<blocks(nc), 256, 0, stream>>>(hout, nc);
        long long ec = (long long)E * cout;
        scatter_edges_kernel<<<blocks(ec), 256, 0, stream>>>(ybuf, basis, widx, ei,
                                                             hout, E, cout, ec);
        norm_deg_kernel<<<blocks(nc), 256, 0, stream>>>(hout, deg, cout, nc);
    };

    sconv(x,  w1,   IN_CH, HID, h1);
    sconv(h1, w2,   HID,   LAT, h2);
    sconv(h2, mu_w, LAT,   LAT, mu);
    sconv(h2, lv_w, LAT,   LAT, logvar);

    // reparameterize + decoder
    long long nl = (long long)N * LAT;
    reparam_kernel<<<blocks(nl), 256, 0, stream>>>(mu, logvar, eps, z, nl);

    dim3 g1((N + GEMM_ROWS - 1) / GEMM_ROWS, HID / 16);
    gemm_wmma_f32_kernel<<<g1, GEMM_THREADS, 0, stream>>>(z, d1_w, d1_b, hdec,
                                                          N, LAT, HID, 1);
    dim3 g2((N + GEMM_ROWS - 1) / GEMM_ROWS, IN_CH / 16);
    gemm_wmma_f32_kernel<<<g2, GEMM_THREADS, 0, stream>>>(hdec, d2_w, d2_b, rec,
                                                          N, HID, IN_CH, 0);
}